// SOMSIRenderer_68667937129123
// MI455X (gfx1250) — compile-verified
//
#include <hip/hip_runtime.h>
#include <math.h>

#define PI_F      3.14159265358979323846f
#define HALF_PI_F 1.57079632679489661923f
#define TWO_PI_F  6.28318530717958647692f

// atan2(y,x) mapped directly into [0, 2*pi), |err| ~1e-5 rad.
__device__ __forceinline__ float fast_atan2_2pi(float y, float x)
{
    float ax = fabsf(x), ay = fabsf(y);
    float mx = fmaxf(ax, ay), mn = fminf(ax, ay);
    float r = mn * __builtin_amdgcn_rcpf(mx);
    if (mx == 0.0f) r = 0.0f;                    // atan2(0,0) = 0
    float s = r * r;
    float p = fmaf(s, fmaf(s, fmaf(s, fmaf(s, 0.0208351f, -0.0851330f),
                                   0.1801410f), -0.3302995f), 0.9998660f);
    p = r * p;                                   // atan(mn/mx) in [0, pi/4]
    if (ay > ax)    p = HALF_PI_F - p;
    if (x < 0.0f)   p = PI_F - p;
    if (y < 0.0f)   p = TWO_PI_F - p;
    return p;
}

// acos(z), z in [-1,1], |err| ~7e-5 rad.
__device__ __forceinline__ float fast_acos(float z)
{
    float xa = fabsf(z);
    float p = fmaf(xa, -0.0187293f, 0.0742610f);
    p = fmaf(p, xa, -0.2121144f);
    p = fmaf(p, xa, 1.5707288f);
    float r = p * __builtin_amdgcn_sqrtf(fmaxf(1.0f - xa, 0.0f));
    return (z < 0.0f) ? PI_F - r : r;
}

template<int H_T, int W_T>
__device__ __forceinline__ void bilinear_setup(float gx, float gy, int H_r, int W_r,
                                               int& o00, int& o10, int& o01, int& o11,
                                               float& w00, float& w10, float& w01, float& w11)
{
    const int H = H_T ? H_T : H_r;
    const int W = W_T ? W_T : W_r;
    // align_corners=False, border padding (clamp indices), matches _grid_sample
    float x = (gx + 1.0f) * 0.5f * (float)W - 0.5f;
    float y = (gy + 1.0f) * 0.5f * (float)H - 0.5f;
    float x0f = floorf(x);
    float y0f = floorf(y);
    float wx = x - x0f;
    float wy = y - y0f;
    int x0 = (int)x0f; x0 = max(0, min(x0, W - 1));
    int x1 = min(x0 + 1, W - 1);
    int y0 = (int)y0f; y0 = max(0, min(y0, H - 1));
    int y1 = min(y0 + 1, H - 1);
    o00 = y0 * W + x0;  o10 = y0 * W + x1;
    o01 = y1 * W + x0;  o11 = y1 * W + x1;
    w00 = (1.0f - wx) * (1.0f - wy);
    w10 = wx * (1.0f - wy);
    w01 = (1.0f - wx) * wy;
    w11 = wx * wy;
}

// Template params == 0 mean "runtime"; nonzero = compile-time constants.
// Grid: (ceil(W/256), H, B)  ->  b, h wave-uniform (SGPR), no divisions anywhere.
template<int S_T, int L_T, int F_T, int H_T, int W_T>
__global__ __launch_bounds__(256)
void somsi_forward(const float* __restrict__ alphas,   // [B,S,1,H,W]
                   const float* __restrict__ feats,    // [B,L,F,H,W]
                   const float* __restrict__ r_mats,   // [B,3,3]
                   const float* __restrict__ t_vecs,   // [B,3]
                   const float* __restrict__ radii,    // [B,S]
                   const float* __restrict__ beta,     // [B,S,L,H,W]
                   float* __restrict__ out,            // [B,F,H,W]
                   int S_r, int L_r, int F_r, int H_r, int W_r)
{
    const int S = S_T ? S_T : S_r;
    const int L = L_T ? L_T : L_r;
    const int F = F_T ? F_T : F_r;
    const int H = H_T ? H_T : H_r;
    const int W = W_T ? W_T : W_r;
    constexpr int LMAX = L_T ? L_T : 8;
    constexpr int FMAX = F_T ? F_T : 16;

    const int w = blockIdx.x * blockDim.x + threadIdx.x;
    const int h = blockIdx.y;     // wave-uniform
    const int b = blockIdx.z;     // wave-uniform
    if (W_T == 0 && w >= W) return;

    const size_t HW = (size_t)H * (size_t)W;
    const int pix = h * W + w;

    // ---- unit ray at equirect pixel center, rotated into source frame ----
    float ph = ((float)h + 0.5f) * (PI_F / (float)H);       // scalar math
    float th = ((float)w + 0.5f) * (TWO_PI_F / (float)W);
    float sph, cph, sth, cth;
    __sincosf(ph, &sph, &cph);
    __sincosf(th, &sth, &cth);
    float rxv = sph * cth, ryv = sph * sth, rzv = cph;

    const float* R = r_mats + b * 9;   // scalar loads (b uniform)
    const float* t = t_vecs + b * 3;
    float dx = fmaf(R[0], rxv, fmaf(R[1], ryv, R[2] * rzv));
    float dy = fmaf(R[3], rxv, fmaf(R[4], ryv, R[5] * rzv));
    float dz = fmaf(R[6], rxv, fmaf(R[7], ryv, R[8] * rzv));
    float tx = t[0], ty = t[1], tz = t[2];

    float a  = fmaf(dx, dx, fmaf(dy, dy, dz * dz));
    float bq = fmaf(dx, tx, fmaf(dy, ty, dz * tz));
    float tt = fmaf(tx, tx, fmaf(ty, ty, tz * tz));
    float inv_a = __builtin_amdgcn_rcpf(a);

    // ---- fused per-sphere warp + sample + over-composite ----
    float T = 1.0f;
    float beta_acc[LMAX];
    #pragma unroll
    for (int l = 0; l < LMAX; ++l) beta_acc[l] = 0.0f;
    float accx = 0.0f, accy = 0.0f;

    const float* aplane = alphas + (size_t)b * S * HW;        // scalar base
    const float* bplane = beta   + (size_t)b * S * L * HW;    // scalar base
    const float* rad    = radii  + (size_t)b * S;             // scalar base

    #pragma unroll 1
    for (int s = 0; s < S; ++s) {
        float r = rad[s];                                      // s_load
        float c = fmaf(-r, r, tt);
        float disc = fmaxf(fmaf(bq, bq, -a * c), 0.0f);
        float tp = (__builtin_amdgcn_sqrtf(disc) - bq) * inv_a;   // far root
        float px = fmaf(tp, dx, tx);
        float py = fmaf(tp, dy, ty);
        float pz = fmaf(tp, dz, tz);
        // theta is scale-invariant: use unnormalized px,py. Only z needs the norm.
        float nrm = __builtin_amdgcn_sqrtf(fmaf(px, px, fmaf(py, py, pz * pz))) + 1e-8f;
        float uz = pz * __builtin_amdgcn_rcpf(nrm);

        float theta = fast_atan2_2pi(py, px);                  // [0, 2pi)
        float phi = fast_acos(fminf(fmaxf(uz, -1.0f), 1.0f));
        float gx = theta * (1.0f / PI_F) - 1.0f;
        float gy = phi * (2.0f / PI_F) - 1.0f;

        int o00, o10, o01, o11; float w00, w10, w01, w11;
        bilinear_setup<H_T, W_T>(gx, gy, H, W, o00, o10, o01, o11, w00, w10, w01, w11);

        const float* ap = aplane + (size_t)s * HW;             // scalar base + vgpr offset
        float aS = fmaf(ap[o00], w00, fmaf(ap[o10], w10,
                   fmaf(ap[o01], w01, ap[o11] * w11)));

        float wT = aS * T;
        const float* bp = bplane + (size_t)s * L * HW;
        #pragma unroll
        for (int l = 0; l < LMAX; ++l) {
            if (L_T == 0 && l >= L) break;
            const float* blp = bp + (size_t)l * HW;
            float v = fmaf(blp[o00], w00, fmaf(blp[o10], w10,
                      fmaf(blp[o01], w01, blp[o11] * w11)));
            beta_acc[l] = fmaf(v, wT, beta_acc[l]);
        }
        accx = fmaf(gx, wT, accx);
        accy = fmaf(gy, wT, accy);
        T *= (1.0f - aS);

        // gfx1250 prefetch: pull next sphere's alpha/beta slabs toward cache.
        if (s + 1 < S) {
            __builtin_prefetch(ap + HW + o00, 0, 1);
            __builtin_prefetch(bp + (size_t)L * HW + o00, 0, 1);
        }
    }

    // ---- sample feats at the composited grid, normalize beta weights ----
    int o00, o10, o01, o11; float w00, w10, w01, w11;
    bilinear_setup<H_T, W_T>(accx, accy, H, W, o00, o10, o01, o11, w00, w10, w01, w11);

    float wsum = 1e-8f;
    #pragma unroll
    for (int l = 0; l < LMAX; ++l) { if (L_T == 0 && l >= L) break; wsum += beta_acc[l]; }
    float inv_ws = __builtin_amdgcn_rcpf(wsum);

    float outv[FMAX];
    #pragma unroll
    for (int f = 0; f < FMAX; ++f) outv[f] = 0.0f;

    const float* fbase = feats + (size_t)b * L * F * HW;       // scalar base
    #pragma unroll
    for (int l = 0; l < LMAX; ++l) {
        if (L_T == 0 && l >= L) break;
        float wl = beta_acc[l] * inv_ws;
        const float* flp = fbase + (size_t)l * F * HW;
        #pragma unroll
        for (int f = 0; f < FMAX; ++f) {
            if (F_T == 0 && f >= F) break;
            const float* fp = flp + (size_t)f * HW;
            float v = fmaf(fp[o00], w00, fmaf(fp[o10], w10,
                      fmaf(fp[o01], w01, fp[o11] * w11)));
            outv[f] = fmaf(wl, v, outv[f]);
        }
    }

    float* op = out + (size_t)b * F * HW + (size_t)pix;
    #pragma unroll
    for (int f = 0; f < FMAX; ++f) {
        if (F_T == 0 && f >= F) break;
        op[(size_t)f * HW] = outv[f];                          // coalesced b32 stores
    }
}

extern "C" void kernel_launch(void* const* d_in, const int* in_sizes, int n_in,
                              void* d_out, int out_size, void* d_ws, size_t ws_size,
                              hipStream_t stream)
{
    const float* alphas = (const float*)d_in[0];  // [B,S,1,H,W]
    const float* feats  = (const float*)d_in[1];  // [B,L,F,H,W]
    const float* r_mats = (const float*)d_in[2];  // [B,3,3]
    const float* t_vecs = (const float*)d_in[3];  // [B,3]
    const float* radii  = (const float*)d_in[4];  // [B,S]
    const float* beta   = (const float*)d_in[5];  // [B,S,L,H,W]
    float* out = (float*)d_out;

    int B = in_sizes[2] / 9;
    int S = in_sizes[4] / B;
    long long HW = (long long)in_sizes[0] / ((long long)B * S);
    int L = (int)((long long)in_sizes[5] / (long long)in_sizes[0]);
    int F = (int)((long long)in_sizes[1] / ((long long)B * L * HW));
    // equirect panorama: W = 2H  =>  HW = 2*H^2
    int H = (int)(sqrt((double)HW * 0.5) + 0.5);
    int W = (int)(HW / H);

    const int block = 256;
    dim3 grid((unsigned)((W + block - 1) / block), (unsigned)H, (unsigned)B);

    if (S == 32 && L == 3 && F == 8 && H == 256 && W == 512) {
        somsi_forward<32, 3, 8, 256, 512><<<grid, block, 0, stream>>>(
            alphas, feats, r_mats, t_vecs, radii, beta, out, S, L, F, H, W);
    } else {
        somsi_forward<0, 0, 0, 0, 0><<<grid, block, 0, stream>>>(
            alphas, feats, r_mats, t_vecs, radii, beta, out, S, L, F, H, W);
    }
}